// BiLevelGraphConvolution_47210280517696
// MI455X (gfx1250) — compile-verified
//
#include <hip/hip_runtime.h>
#include <hip/hip_bf16.h>

#define NS_ 50000
#define NT_ 50000
#define NN_ 100000
#define DD_ 128
#define NEG_SLOPE 0.2f

typedef __attribute__((ext_vector_type(2))) float v2f;
typedef __attribute__((ext_vector_type(8))) float v8f;

__device__ __forceinline__ float leaky(float x) { return x > 0.f ? x : NEG_SLOPE * x; }
// monotonic unsigned key for float atomicMax
__device__ __forceinline__ unsigned fkey(float f) {
    unsigned u = __float_as_uint(f);
    return (u >> 31) ? ~u : (u | 0x80000000u);
}
__device__ __forceinline__ float fdec(unsigned k) {
    return (k >> 31) ? __uint_as_float(k & 0x7fffffffu) : __uint_as_float(~k);
}

// One wave computes Y[yrow0:yrow0+16, 0:128] = X[xrow0:xrow0+16, :] @ W^T (+bias)
// W is [128,128] row-major (out x in). Uses V_WMMA_F32_16X16X4_F32, K-chained x32.
__device__ __forceinline__ void gemm16rows(const float* __restrict__ X, int xrow0,
                                           const float* __restrict__ W,
                                           const float* __restrict__ bias,
                                           float* __restrict__ Y, int yrow0) {
    const int lane  = threadIdx.x & 31;
    const int rsel  = lane & 15;   // row within 16 / col within 16
    const int khalf = lane >> 4;   // K half selector

    // A tile 16x4 f32 layout: lanes0-15 VGPR0=K0,VGPR1=K1 ; lanes16-31 VGPR0=K2,VGPR1=K3
    v2f a[32];
    const float* xrow = X + (size_t)(xrow0 + rsel) * DD_ + 2 * khalf;
#pragma unroll
    for (int ks = 0; ks < 32; ++ks) a[ks] = *(const v2f*)(xrow + ks * 4);

#pragma unroll
    for (int nt = 0; nt < 8; ++nt) {
        const int n0 = nt * 16;
        const float* wrow = W + (size_t)(n0 + rsel) * DD_ + 2 * khalf;  // B = W^T tile
        v8f c = {0.f, 0.f, 0.f, 0.f, 0.f, 0.f, 0.f, 0.f};
#pragma unroll
        for (int ks = 0; ks < 32; ++ks) {
            v2f b = *(const v2f*)(wrow + ks * 4);
            c = __builtin_amdgcn_wmma_f32_16x16x4_f32(false, a[ks], false, b,
                                                      (short)0, c, false, false);
        }
        const float bv = bias ? bias[n0 + rsel] : 0.f;
        // C/D layout: VGPR r -> M = r + 8*khalf, N = n0 + rsel
#pragma unroll
        for (int r = 0; r < 8; ++r)
            Y[(size_t)(yrow0 + r + 8 * khalf) * DD_ + n0 + rsel] = c[r] + bv;
    }
}

// x[0:NS] = x_s @ Ws^T + bs ; x[NS:NN] = x_t @ Wt^T + bt   (NS % 16 == 0)
__global__ void k_linear(const float* __restrict__ xs, const float* __restrict__ Ws,
                         const float* __restrict__ bs, const float* __restrict__ xt,
                         const float* __restrict__ Wt, const float* __restrict__ bt,
                         float* __restrict__ x) {
    int wave = blockIdx.x * (blockDim.x >> 5) + (threadIdx.x >> 5);
    int m0 = wave * 16;
    if (m0 >= NN_) return;
    if (m0 < NS_) gemm16rows(xs, m0, Ws, bs, x, m0);
    else          gemm16rows(xt, m0 - NS_, Wt, bt, x, m0);
}

__global__ void k_gemm(const float* __restrict__ x, const float* __restrict__ W,
                       float* __restrict__ h) {
    int wave = blockIdx.x * (blockDim.x >> 5) + (threadIdx.x >> 5);
    int m0 = wave * 16;
    if (m0 >= NN_) return;
    gemm16rows(x, m0, W, nullptr, h, m0);
}

// per-node attention scalars: a1s_h = h1·a1s, etc. One wave per node.
__global__ void k_dot(const float* __restrict__ h1, const float* __restrict__ h2,
                      const float* __restrict__ a1s, const float* __restrict__ a1d,
                      const float* __restrict__ a2s, const float* __restrict__ a2d,
                      float* __restrict__ o1s, float* __restrict__ o1d,
                      float* __restrict__ o2s, float* __restrict__ o2d) {
    int node = blockIdx.x * (blockDim.x >> 5) + (threadIdx.x >> 5);
    if (node >= NN_) return;
    int lane = threadIdx.x & 31;
    float4 v1 = *(const float4*)(h1 + (size_t)node * DD_ + lane * 4);
    float4 v2 = *(const float4*)(h2 + (size_t)node * DD_ + lane * 4);
    float4 w1s = *(const float4*)(a1s + lane * 4);
    float4 w1d = *(const float4*)(a1d + lane * 4);
    float4 w2s = *(const float4*)(a2s + lane * 4);
    float4 w2d = *(const float4*)(a2d + lane * 4);
    float s0 = v1.x * w1s.x + v1.y * w1s.y + v1.z * w1s.z + v1.w * w1s.w;
    float s1 = v1.x * w1d.x + v1.y * w1d.y + v1.z * w1d.z + v1.w * w1d.w;
    float s2 = v2.x * w2s.x + v2.y * w2s.y + v2.z * w2s.z + v2.w * w2s.w;
    float s3 = v2.x * w2d.x + v2.y * w2d.y + v2.z * w2d.z + v2.w * w2d.w;
#pragma unroll
    for (int off = 16; off >= 1; off >>= 1) {
        s0 += __shfl_xor(s0, off, 32);
        s1 += __shfl_xor(s1, off, 32);
        s2 += __shfl_xor(s2, off, 32);
        s3 += __shfl_xor(s3, off, 32);
    }
    if (lane == 0) { o1s[node] = s0; o1d[node] = s1; o2s[node] = s2; o2d[node] = s3; }
}

__global__ void k_zero(float4* __restrict__ p, int n4) {
    int i = blockIdx.x * blockDim.x + threadIdx.x;
    if (i < n4) p[i] = make_float4(0.f, 0.f, 0.f, 0.f);
}

// seed emax with self-loop score, zero denom. Node i<NS belongs to conv2, else conv1.
__global__ void k_init(const float* __restrict__ o1s, const float* __restrict__ o1d,
                       const float* __restrict__ o2s, const float* __restrict__ o2d,
                       unsigned* __restrict__ emax, float* __restrict__ den) {
    int i = blockIdx.x * blockDim.x + threadIdx.x;
    if (i >= NN_) return;
    float e = (i < NS_) ? leaky(o2s[i] + o2d[i]) : leaky(o1s[i] + o1d[i]);
    emax[i] = fkey(e);
    den[i] = 0.f;
}

__global__ void k_edge_max(const int* __restrict__ s1, int s1add,
                           const int* __restrict__ d1, int d1add,
                           const int* __restrict__ s2, const int* __restrict__ d2, int E,
                           const float* __restrict__ as, const float* __restrict__ ad,
                           unsigned* __restrict__ emax, int keep_lo, int keep_hi) {
    int t = blockIdx.x * blockDim.x + threadIdx.x;
    if (t >= 2 * E) return;
    int src, dst;
    if (t < E) { src = s1[t] + s1add; dst = d1[t] + d1add; }
    else       { src = s2[t - E];     dst = d2[t - E]; }
    if (dst < keep_lo || dst >= keep_hi) return;
    float e = leaky(as[src] + ad[dst]);
    atomicMax(emax + dst, fkey(e));
}

__global__ void k_edge_sum(const int* __restrict__ s1, int s1add,
                           const int* __restrict__ d1, int d1add,
                           const int* __restrict__ s2, const int* __restrict__ d2, int E,
                           const float* __restrict__ as, const float* __restrict__ ad,
                           const unsigned* __restrict__ emax, float* __restrict__ den,
                           int keep_lo, int keep_hi) {
    int t = blockIdx.x * blockDim.x + threadIdx.x;
    if (t >= 2 * E) return;
    int src, dst;
    if (t < E) { src = s1[t] + s1add; dst = d1[t] + d1add; }
    else       { src = s2[t - E];     dst = d2[t - E]; }
    if (dst < keep_lo || dst >= keep_hi) return;
    float e = leaky(as[src] + ad[dst]);
    float w = __expf(e - fdec(emax[dst]));
    __hip_atomic_fetch_add(den + dst, w, __ATOMIC_RELAXED, __HIP_MEMORY_SCOPE_AGENT);
}

// add self-loop exp to denom (exclusive writer per node in this launch)
__global__ void k_node_sum(const float* __restrict__ o1s, const float* __restrict__ o1d,
                           const float* __restrict__ o2s, const float* __restrict__ o2d,
                           const unsigned* __restrict__ emax, float* __restrict__ den) {
    int i = blockIdx.x * blockDim.x + threadIdx.x;
    if (i >= NN_) return;
    float e = (i < NS_) ? leaky(o2s[i] + o2d[i]) : leaky(o1s[i] + o1d[i]);
    den[i] += __expf(e - fdec(emax[i]));
}

// one wave per edge: out[dst] += alpha * h[src]  (float4 per lane, f32 atomics)
__global__ void k_scatter(const int* __restrict__ s1, int s1add,
                          const int* __restrict__ d1, int d1add,
                          const int* __restrict__ s2, const int* __restrict__ d2, int E,
                          const float* __restrict__ as, const float* __restrict__ ad,
                          const unsigned* __restrict__ emax, const float* __restrict__ den,
                          const float* __restrict__ h, float* __restrict__ outbase,
                          int keep_lo, int keep_hi) {
    int wave = blockIdx.x * (blockDim.x >> 5) + (threadIdx.x >> 5);
    if (wave >= 2 * E) return;
    int lane = threadIdx.x & 31;
    int src, dst;
    if (wave < E) { src = s1[wave] + s1add; dst = d1[wave] + d1add; }
    else          { src = s2[wave - E];     dst = d2[wave - E]; }
    if (dst < keep_lo || dst >= keep_hi) return;
    float e = leaky(as[src] + ad[dst]);
    float alpha = __expf(e - fdec(emax[dst])) / (den[dst] + 1e-16f);
    float4 hv = *(const float4*)(h + (size_t)src * DD_ + lane * 4);
    float* orow = outbase + (size_t)(dst - keep_lo) * DD_ + lane * 4;
    __hip_atomic_fetch_add(orow + 0, hv.x * alpha, __ATOMIC_RELAXED, __HIP_MEMORY_SCOPE_AGENT);
    __hip_atomic_fetch_add(orow + 1, hv.y * alpha, __ATOMIC_RELAXED, __HIP_MEMORY_SCOPE_AGENT);
    __hip_atomic_fetch_add(orow + 2, hv.z * alpha, __ATOMIC_RELAXED, __HIP_MEMORY_SCOPE_AGENT);
    __hip_atomic_fetch_add(orow + 3, hv.w * alpha, __ATOMIC_RELAXED, __HIP_MEMORY_SCOPE_AGENT);
}

// add self-loop contribution + ReLU. Output row for node i is out + i*128 for both convs.
__global__ void k_final(const float* __restrict__ h1, const float* __restrict__ h2,
                        const float* __restrict__ o1s, const float* __restrict__ o1d,
                        const float* __restrict__ o2s, const float* __restrict__ o2d,
                        const unsigned* __restrict__ emax, const float* __restrict__ den,
                        float* __restrict__ out) {
    int node = blockIdx.x * (blockDim.x >> 5) + (threadIdx.x >> 5);
    if (node >= NN_) return;
    int lane = threadIdx.x & 31;
    float e; const float* h;
    if (node < NS_) { e = leaky(o2s[node] + o2d[node]); h = h2; }
    else            { e = leaky(o1s[node] + o1d[node]); h = h1; }
    float alpha = __expf(e - fdec(emax[node])) / (den[node] + 1e-16f);
    float4 hv = *(const float4*)(h + (size_t)node * DD_ + lane * 4);
    float* orow = out + (size_t)node * DD_ + lane * 4;
    float4 o = *(float4*)orow;
    o.x = fmaxf(o.x + hv.x * alpha, 0.f);
    o.y = fmaxf(o.y + hv.y * alpha, 0.f);
    o.z = fmaxf(o.z + hv.z * alpha, 0.f);
    o.w = fmaxf(o.w + hv.w * alpha, 0.f);
    *(float4*)orow = o;
}

extern "C" void kernel_launch(void* const* d_in, const int* in_sizes, int n_in,
                              void* d_out, int out_size, void* d_ws, size_t ws_size,
                              hipStream_t stream) {
    (void)n_in; (void)out_size; (void)ws_size;
    const int* ei0 = (const int*)d_in[0];
    const int E = in_sizes[0] / 2;  // 500000
    const int* ei1 = ei0 + E;
    const int* pe0 = (const int*)d_in[1];
    const int* pe1 = pe0 + E;
    const int* ae0 = (const int*)d_in[2];
    const int* ae1 = ae0 + E;
    const float* xs = (const float*)d_in[3];
    const float* xt = (const float*)d_in[4];
    const float* Ws = (const float*)d_in[5];
    const float* bs = (const float*)d_in[6];
    const float* Wt = (const float*)d_in[7];
    const float* bt = (const float*)d_in[8];
    const float* W1 = (const float*)d_in[9];
    const float* a1s = (const float*)d_in[10];
    const float* a1d = (const float*)d_in[11];
    const float* W2 = (const float*)d_in[12];
    const float* a2s = (const float*)d_in[13];
    const float* a2d = (const float*)d_in[14];
    float* out = (float*)d_out;

    // workspace carve-up (~156 MB)
    char* ws = (char*)d_ws;
    float* x  = (float*)ws;  ws += (size_t)NN_ * DD_ * sizeof(float);
    float* h1 = (float*)ws;  ws += (size_t)NN_ * DD_ * sizeof(float);
    float* h2 = (float*)ws;  ws += (size_t)NN_ * DD_ * sizeof(float);
    float* o1s = (float*)ws; ws += (size_t)NN_ * sizeof(float);
    float* o1d = (float*)ws; ws += (size_t)NN_ * sizeof(float);
    float* o2s = (float*)ws; ws += (size_t)NN_ * sizeof(float);
    float* o2d = (float*)ws; ws += (size_t)NN_ * sizeof(float);
    unsigned* emax = (unsigned*)ws; ws += (size_t)NN_ * sizeof(unsigned);
    float* den = (float*)ws;

    const dim3 B(256);
    const int tiles = NN_ / 16;                    // 6250 row tiles
    const int gemmBlocks = (tiles + 7) / 8;        // 8 waves/block
    const int nodeWaveBlocks = (NN_ + 7) / 8;      // wave-per-node kernels
    const int nodeBlocks = (NN_ + 255) / 256;
    const int edgeBlocks = (2 * E + 255) / 256;
    const int scatBlocks = (2 * E + 7) / 8;        // wave-per-edge

    k_linear<<<gemmBlocks, B, 0, stream>>>(xs, Ws, bs, xt, Wt, bt, x);
    k_gemm<<<gemmBlocks, B, 0, stream>>>(x, W1, h1);
    k_gemm<<<gemmBlocks, B, 0, stream>>>(x, W2, h2);
    k_dot<<<nodeWaveBlocks, B, 0, stream>>>(h1, h2, a1s, a1d, a2s, a2d, o1s, o1d, o2s, o2d);
    k_zero<<<(NN_ * DD_ / 4 + 255) / 256, B, 0, stream>>>((float4*)out, NN_ * DD_ / 4);
    k_init<<<nodeBlocks, B, 0, stream>>>(o1s, o1d, o2s, o2d, emax, den);
    // conv1: edges (src, dst+NS) ++ author, keep dst in [NS, NN)
    k_edge_max<<<edgeBlocks, B, 0, stream>>>(ei0, 0, ei1, NS_, ae0, ae1, E, o1s, o1d, emax, NS_, NN_);
    // conv2: edges (dst+NS, src) ++ paper, keep dst in [0, NS)
    k_edge_max<<<edgeBlocks, B, 0, stream>>>(ei1, NS_, ei0, 0, pe0, pe1, E, o2s, o2d, emax, 0, NS_);
    k_node_sum<<<nodeBlocks, B, 0, stream>>>(o1s, o1d, o2s, o2d, emax, den);
    k_edge_sum<<<edgeBlocks, B, 0, stream>>>(ei0, 0, ei1, NS_, ae0, ae1, E, o1s, o1d, emax, den, NS_, NN_);
    k_edge_sum<<<edgeBlocks, B, 0, stream>>>(ei1, NS_, ei0, 0, pe0, pe1, E, o2s, o2d, emax, den, 0, NS_);
    k_scatter<<<scatBlocks, B, 0, stream>>>(ei0, 0, ei1, NS_, ae0, ae1, E, o1s, o1d, emax, den,
                                            h1, out + (size_t)NS_ * DD_, NS_, NN_);
    k_scatter<<<scatBlocks, B, 0, stream>>>(ei1, NS_, ei0, 0, pe0, pe1, E, o2s, o2d, emax, den,
                                            h2, out, 0, NS_);
    k_final<<<nodeWaveBlocks, B, 0, stream>>>(h1, h2, o1s, o1d, o2s, o2d, emax, den, out);
}